// LearnableLaplaceTrans_43267500540575
// MI455X (gfx1250) — compile-verified
//
#include <hip/hip_runtime.h>
#include <hip/hip_bf16.h>
#include <math.h>

typedef __attribute__((ext_vector_type(2))) float v2f;
typedef __attribute__((ext_vector_type(8))) float v8f;

#define BB   16
#define TT   4096
#define FF   128
#define KK   64
#define NC   64          // chunks per sequence
#define CL   64          // chunk length (NC*CL == TT)
#define PI_F 3.14159265358979323846f

// ---------------- workspace layout (in floats) ----------------
#define WN_OFF   0                          // Wn: 64*128
#define PAR_OFF  (WN_OFF + KK*FF)           // alpha[64], omega[64], inv_den[64], beff[64]
#define U_OFF    (PAR_OFF + 4*KK)           // u_eff: B*T*K
#define AGG_OFF  (U_OFF + (size_t)BB*TT*KK) // aggregates: B*K*NC*4

__device__ __forceinline__ float softplusf(float x) {
    return (x > 0.f) ? (x + log1pf(__expf(-x))) : log1pf(__expf(x));
}

__device__ __forceinline__ float block_reduce_sum(float v, float* red) {
    int tid = threadIdx.x;
    red[tid] = v;
    __syncthreads();
    for (int s = 64; s > 0; s >>= 1) {
        if (tid < s) red[tid] += red[tid + s];
        __syncthreads();
    }
    float r = red[0];
    __syncthreads();
    return r;
}

// ---------------------------------------------------------------
// Kernel 0: spectral norm (20 power iterations) + scalar params.
// One block of 128 threads; W (32KB) lives in LDS.
// ---------------------------------------------------------------
__global__ void llt_setup_kernel(const float* __restrict__ W,
                                 const float* __restrict__ s_real_raw,
                                 const float* __restrict__ s_imag,
                                 const float* __restrict__ tau,
                                 const float* __restrict__ b_param,
                                 float* __restrict__ ws) {
    __shared__ float sW[KK * FF];
    __shared__ float su[KK];
    __shared__ float sv[FF];
    __shared__ float red[128];

    const int tid = threadIdx.x;
    for (int i = tid; i < KK * FF; i += 128) sW[i] = W[i];
    if (tid < KK) su[tid] = 0.125f;            // 1/sqrt(64)
    __syncthreads();

    for (int it = 0; it < 20; ++it) {
        // v = normalize(W^T u) : thread tid owns feature f = tid
        float acc = 0.f;
        for (int k = 0; k < KK; ++k) acc += sW[k * FF + tid] * su[k];
        float nv = sqrtf(block_reduce_sum(acc * acc, red)) + 1e-12f;
        sv[tid] = acc / nv;
        __syncthreads();

        // u = normalize(W v) : threads 0..63 own mode k = tid
        float au = 0.f;
        if (tid < KK) {
            for (int f = 0; f < FF; ++f) au += sW[tid * FF + f] * sv[f];
        }
        float nu = sqrtf(block_reduce_sum((tid < KK) ? au * au : 0.f, red)) + 1e-12f;
        if (tid < KK) su[tid] = au / nu;
        __syncthreads();
    }

    // sigma = u . (W v)
    float au = 0.f;
    if (tid < KK) {
        for (int f = 0; f < FF; ++f) au += sW[tid * FF + f] * sv[f];
    }
    float sigma = block_reduce_sum((tid < KK) ? au * su[tid] : 0.f, red);
    float inv_sigma = 1.0f / sigma;

    float* Wn = ws + WN_OFF;
    for (int i = tid; i < KK * FF; i += 128) Wn[i] = sW[i] * inv_sigma;

    if (tid < KK) {
        float tau_sp = softplusf(tau[0]) + 0.001f;
        float alpha  = (softplusf(s_real_raw[tid]) + 1e-6f) * tau_sp;
        float omega  = fminf(fmaxf(s_imag[tid] * tau_sp, -PI_F), PI_F);
        float den    = fmaxf(alpha * alpha + omega * omega, 1e-6f);
        float* par = ws + PAR_OFF;
        par[tid]            = alpha;
        par[KK + tid]       = omega;
        par[2 * KK + tid]   = 1.0f / den;
        par[3 * KK + tid]   = softplusf(b_param[tid]) + 1e-8f;
    }
}

// ---------------------------------------------------------------
// Kernel 1: u_eff = (x @ Wn^T + b_lin) * b_eff  via f32 WMMA.
// Block = 128 threads (4 waves); block owns 64 rows (4 M-tiles).
// Wave w owns output columns [16w,16w+16) and holds its 32 B-fragments
// (Wn) in registers, reused across all 4 M-tiles -> 128 WMMAs per wave
// for 32 B loads. A-tile staged in LDS, rows padded to 132 floats so
// the column-wise b64 fragment reads hit all 64 banks exactly once.
// ---------------------------------------------------------------
#define XPAD   132
#define MROWS  64          // rows per block (4 M-tiles of 16)

__global__ void llt_gemm_kernel(const float* __restrict__ x,
                                const float* __restrict__ b_lin,
                                const float* __restrict__ ws_ro,
                                float* __restrict__ u_eff) {
    __shared__ float sx[MROWS * XPAD];

    const int tid = threadIdx.x;
    const int m0  = blockIdx.x * MROWS;

    // stage 64x128 f32 A-tile into LDS: each thread copies half a row (16 float4s)
    {
        const int row  = tid >> 1;
        const int half = tid & 1;
        const float4* s4 = (const float4*)(x + (size_t)(m0 + row) * FF + half * 64);
        float4* d4 = (float4*)(sx + row * XPAD + half * 64);
#pragma unroll
        for (int q = 0; q < 16; ++q) d4[q] = s4[q];
    }
    __syncthreads();

    const int lane = tid & 31;
    const int wv   = tid >> 5;          // N-tile index 0..3
    const int ll   = lane & 15;
    const int lh   = lane >> 4;
    const int n    = wv * 16 + ll;      // output mode column

    // Hoist all 32 B-fragments for this wave's N-tile into registers.
    const float* wrow = ws_ro + WN_OFF + (size_t)n * FF;
    v2f barr[32];
#pragma unroll
    for (int s = 0; s < 32; ++s) {
        const int c = 4 * s + 2 * lh;
        barr[s].x = wrow[c];
        barr[s].y = wrow[c + 1];
    }

    const float blv = b_lin[n];
    const float bev = ws_ro[PAR_OFF + 3 * KK + n];

    for (int mt = 0; mt < 4; ++mt) {
        const float* xrow = sx + (mt * 16 + ll) * XPAD;
        v8f acc = {};
#pragma unroll
        for (int s = 0; s < 32; ++s) {
            const int c = 4 * s + 2 * lh;
            v2f a;
            a.x = xrow[c];
            a.y = xrow[c + 1];
            acc = __builtin_amdgcn_wmma_f32_16x16x4_f32(
                /*neg_a=*/false, a, /*neg_b=*/false, barr[s],
                /*c_mod=*/(short)0, acc, /*reuse_a=*/false, /*reuse_b=*/false);
        }
#pragma unroll
        for (int v = 0; v < 8; ++v) {
            const int m = m0 + mt * 16 + v + 8 * lh;
            u_eff[(size_t)m * KK + n] = (acc[v] + blv) * bev;
        }
    }
}

// ---------------------------------------------------------------
// Scan step helper: given dt, per-mode params and u, produce the
// complex transition z = rho*e^{i theta} and input w = psi * u.
// ---------------------------------------------------------------
__device__ __forceinline__ void llt_step_coeffs(float d, float alpha, float omega,
                                                float invden, float u,
                                                float& zc, float& zs,
                                                float& wc, float& wsi) {
    float rho = __expf(-alpha * d);
    float st, ct;
    __sincosf(omega * d, &st, &ct);
    zc = rho * ct;
    zs = rho * st;
    float pc = (-omega + rho * (alpha * st + omega * ct)) * invden;
    float ps = ( alpha - rho * (alpha * ct - omega * st)) * invden;
    wc  = pc * u;
    wsi = ps * u;
}

// ---------------------------------------------------------------
// Kernel 2: per-chunk local scan -> aggregates (Z = prod z, H = local h).
// grid (NC, B), block 64 (one thread per mode k).
// ---------------------------------------------------------------
__global__ void llt_scan_local_kernel(const float* __restrict__ dt,
                                      const float* __restrict__ ws_ro,
                                      float* __restrict__ agg) {
    const int k     = threadIdx.x;
    const int chunk = blockIdx.x;
    const int b     = blockIdx.y;

    const float alpha  = ws_ro[PAR_OFF + k];
    const float omega  = ws_ro[PAR_OFF + KK + k];
    const float invden = ws_ro[PAR_OFF + 2 * KK + k];

    const int t0 = chunk * CL;
    const float* dtp = dt + (size_t)b * TT + t0;
    const float* up  = ws_ro + U_OFF + ((size_t)b * TT + t0) * KK + k;

    float Zc = 1.f, Zs = 0.f, Hc = 0.f, Hs = 0.f;
    for (int j = 0; j < CL; ++j) {
        float zc, zs, wc, wsv;
        llt_step_coeffs(dtp[j], alpha, omega, invden, up[(size_t)j * KK],
                        zc, zs, wc, wsv);
        float hc = zc * Hc - zs * Hs + wc;
        float hs = zc * Hs + zs * Hc + wsv;
        Hc = hc; Hs = hs;
        float nzc = zc * Zc - zs * Zs;
        float nzs = zc * Zs + zs * Zc;
        Zc = nzc; Zs = nzs;
    }
    const size_t idx = (((size_t)b * KK + k) * NC + chunk) * 4;
    agg[idx]     = Zc;
    agg[idx + 1] = Zs;
    agg[idx + 2] = Hc;
    agg[idx + 3] = Hs;
}

// ---------------------------------------------------------------
// Kernel 3: exclusive prefix over the NC chunk aggregates.
// grid B, block 64. Overwrites agg[..+2/+3] with chunk START states.
// ---------------------------------------------------------------
__global__ void llt_scan_prefix_kernel(float* __restrict__ agg) {
    const int k = threadIdx.x;
    const int b = blockIdx.x;
    const size_t base = ((size_t)b * KK + k) * NC * 4;

    float hc = 0.f, hs = 0.f;
    for (int c = 0; c < NC; ++c) {
        const size_t i = base + (size_t)c * 4;
        float zc = agg[i], zs = agg[i + 1];
        float lc = agg[i + 2], ls = agg[i + 3];
        agg[i + 2] = hc;            // start state of chunk c
        agg[i + 3] = hs;
        float nhc = zc * hc - zs * hs + lc;
        float nhs = zc * hs + zs * hc + ls;
        hc = nhc; hs = nhs;
    }
}

// ---------------------------------------------------------------
// Kernel 4: re-run each chunk from its start state and emit C,S.
// grid (NC, B), block 64.
// ---------------------------------------------------------------
__global__ void llt_scan_emit_kernel(const float* __restrict__ dt,
                                     const float* __restrict__ ws_ro,
                                     float* __restrict__ out) {
    const int k     = threadIdx.x;
    const int chunk = blockIdx.x;
    const int b     = blockIdx.y;

    const float alpha  = ws_ro[PAR_OFF + k];
    const float omega  = ws_ro[PAR_OFF + KK + k];
    const float invden = ws_ro[PAR_OFF + 2 * KK + k];

    const int t0 = chunk * CL;
    const float* dtp = dt + (size_t)b * TT + t0;
    const float* up  = ws_ro + U_OFF + ((size_t)b * TT + t0) * KK + k;
    const float* agg = ws_ro + AGG_OFF;
    const size_t aidx = (((size_t)b * KK + k) * NC + chunk) * 4;

    float hc = agg[aidx + 2];
    float hs = agg[aidx + 3];

    float* op = out + ((size_t)b * TT + t0) * (2 * KK) + k;
    for (int j = 0; j < CL; ++j) {
        float zc, zs, wc, wsv;
        llt_step_coeffs(dtp[j], alpha, omega, invden, up[(size_t)j * KK],
                        zc, zs, wc, wsv);
        float nhc = zc * hc - zs * hs + wc;
        float nhs = zc * hs + zs * hc + wsv;
        hc = nhc; hs = nhs;
        op[(size_t)j * (2 * KK)]      = hc;   // C at [..., k]
        op[(size_t)j * (2 * KK) + KK] = hs;   // S at [..., K+k]
    }
}

// ---------------------------------------------------------------
extern "C" void kernel_launch(void* const* d_in, const int* in_sizes, int n_in,
                              void* d_out, int out_size, void* d_ws, size_t ws_size,
                              hipStream_t stream) {
    (void)in_sizes; (void)n_in; (void)out_size; (void)ws_size;

    const float* x          = (const float*)d_in[0];   // [B,T,F]
    const float* dt         = (const float*)d_in[1];   // [B,T]
    const float* s_real_raw = (const float*)d_in[2];   // [K]
    const float* s_imag     = (const float*)d_in[3];   // [K]
    const float* tau        = (const float*)d_in[4];   // [1]
    const float* W          = (const float*)d_in[5];   // [K,F]
    const float* b_lin      = (const float*)d_in[6];   // [K]
    const float* b_param    = (const float*)d_in[7];   // [K]

    float* ws  = (float*)d_ws;
    float* out = (float*)d_out;

    // 0: spectral norm + params
    llt_setup_kernel<<<1, 128, 0, stream>>>(W, s_real_raw, s_imag, tau, b_param, ws);

    // 1: WMMA GEMM -> u_eff (64 rows per block)
    llt_gemm_kernel<<<(BB * TT) / MROWS, 128, 0, stream>>>(x, b_lin, ws, ws + U_OFF);

    // 2: chunk-local scans
    llt_scan_local_kernel<<<dim3(NC, BB), 64, 0, stream>>>(dt, ws, ws + AGG_OFF);

    // 3: prefix over chunks
    llt_scan_prefix_kernel<<<BB, 64, 0, stream>>>(ws + AGG_OFF);

    // 4: emit outputs
    llt_scan_emit_kernel<<<dim3(NC, BB), 64, 0, stream>>>(dt, ws, out);
}